// UHGMultiHeadAttention_73203422593573
// MI455X (gfx1250) — compile-verified
//
#include <hip/hip_runtime.h>

// UHG multi-head attention, fused for MI455X (gfx1250, wave32, WMMA f32 16x16x4).
// B=8, N=1024, D=512, H=8. Outputs: out [B,N,512] then attn [B,H,N,N] in d_out.

#define B_  8
#define N_  1024
#define D_  512
#define H_  8
#define DP_ 513
#define EPSF 1e-6f

#define QB  32     // q rows per attention block
#define PAD 1028   // LDS row stride (floats); rows land on distinct banks

typedef __attribute__((ext_vector_type(2))) float v2f;
typedef __attribute__((ext_vector_type(8))) float v8f;

__device__ __forceinline__ v8f wmma_f32_16x16x4(v2f a, v2f b, v8f c) {
  // (neg_a, A, neg_b, B, c_mod, C, reuse_a, reuse_b)
  return __builtin_amdgcn_wmma_f32_16x16x4_f32(false, a, false, b, (short)0, c,
                                               false, false);
}

// ---------------------------------------------------------------------------
// Kernel 1: per (b,h,n) compute normalized q0,q1,k0,k1 (only components used
// by the cross-ratio scores). One wave per (b,h,n); lane-parallel dot products.
// ---------------------------------------------------------------------------
__global__ void qk01_kernel(const float* __restrict__ query,
                            const float* __restrict__ key,
                            const float* __restrict__ Wq,
                            const float* __restrict__ Wk,
                            float* __restrict__ q0a, float* __restrict__ q1a,
                            float* __restrict__ k0a, float* __restrict__ k1a,
                            float fqk) {
  const int wid  = (blockIdx.x * blockDim.x + threadIdx.x) >> 5; // (b*H+h)*N+n
  const int lane = threadIdx.x & 31;
  const int b = wid >> 13;            // H*N = 8192 = 2^13
  const int h = (wid >> 10) & (H_ - 1);
  const int n = wid & (N_ - 1);

  const float* qrow = query + ((size_t)b * N_ + n) * D_;
  const float* krow = key   + ((size_t)b * N_ + n) * D_;
  const float* wq = Wq + (size_t)h * DP_ * DP_;   // rows 0 and 1, cols 0..511
  const float* wk = Wk + (size_t)h * DP_ * DP_;

  float aq0 = 0.f, aq1 = 0.f, ak0 = 0.f, ak1 = 0.f;
#pragma unroll
  for (int i = 0; i < D_ / 32; ++i) {
    const int d = lane + 32 * i;
    const float xq = qrow[d], xk = krow[d];
    aq0 += xq * wq[d];        aq1 += xq * wq[DP_ + d];
    ak0 += xk * wk[d];        ak1 += xk * wk[DP_ + d];
  }
#pragma unroll
  for (int off = 16; off >= 1; off >>= 1) {
    aq0 += __shfl_xor(aq0, off, 32);
    aq1 += __shfl_xor(aq1, off, 32);
    ak0 += __shfl_xor(ak0, off, 32);
    ak1 += __shfl_xor(ak1, off, 32);
  }
  if (lane == 0) {
    q0a[wid] = fqk * aq0;  q1a[wid] = fqk * aq1;
    k0a[wid] = fqk * ak0;  k1a[wid] = fqk * ak1;
  }
}

// ---------------------------------------------------------------------------
// Kernel 2: C[m,e] = alpha * sum_k A[m,k] * W[e,k]   (K = 512, 512 cols)
// Compile-time strides: lda=512, ldw=513, ldc=512 -> all loads are immediate
// offsets off two base pointers. Block: 8 waves -> 128 rows x 128 cols.
// Wave: 16x128 (8 tiles), ping-pong double buffered. blockIdx.z = head.
// ---------------------------------------------------------------------------
__global__ void __launch_bounds__(256, 1)
gemm_xwT_kernel(const float* __restrict__ A, long a_hs,
                const float* __restrict__ W, long w_hs,
                float* __restrict__ C, long c_hs,
                float alpha) {
  constexpr int LDA = D_, LDW = DP_, LDC = D_;

  const float* Ah = A + (size_t)blockIdx.z * a_hs;
  const float* Wh = W + (size_t)blockIdx.z * w_hs;
  float* Ch = C + (size_t)blockIdx.z * c_hs;

  const int wave = threadIdx.x >> 5;
  const int lane = threadIdx.x & 31;
  const int m0 = blockIdx.x * 128 + wave * 16;
  const int e0 = blockIdx.y * 128;
  const int khalf = 2 * (lane >> 4);   // fragment holds K = kb+khalf, kb+khalf+1

  const float* arow  = Ah + (size_t)(m0 + (lane & 15)) * LDA + khalf;
  const float* wbase = Wh + (size_t)(e0 + (lane & 15)) * LDW + khalf;

  auto loadA = [&](int kb) {
    v2f a; a.x = arow[kb]; a.y = arow[kb + 1]; return a;
  };
  auto loadB = [&](int kb, int t) {
    const float* wp = wbase + t * (16 * LDW) + kb;   // immediate offsets
    v2f b; b.x = wp[0]; b.y = wp[1];                 // B[k][n] = W[n][k]
    return b;
  };

  v8f acc[8] = {};
  v2f aA = loadA(0);
  v2f bA[8];
#pragma unroll
  for (int t = 0; t < 8; ++t) bA[t] = loadB(0, t);
  v2f aB;
  v2f bB[8];

#pragma unroll 1
  for (int kb = 0; kb < D_ - 8; kb += 8) {
    aB = loadA(kb + 4);
#pragma unroll
    for (int t = 0; t < 8; ++t) bB[t] = loadB(kb + 4, t);
    if ((kb & 63) == 0)
      __builtin_prefetch(arow + kb + 128, 0, 0);     // global_prefetch_b8
#pragma unroll
    for (int t = 0; t < 8; ++t) acc[t] = wmma_f32_16x16x4(aA, bA[t], acc[t]);
    aA = loadA(kb + 8);
#pragma unroll
    for (int t = 0; t < 8; ++t) bA[t] = loadB(kb + 8, t);
#pragma unroll
    for (int t = 0; t < 8; ++t) acc[t] = wmma_f32_16x16x4(aB, bB[t], acc[t]);
  }
  // epilogue: compute K-steps D_-8 (in A set) and D_-4 (loaded here)
  aB = loadA(D_ - 4);
#pragma unroll
  for (int t = 0; t < 8; ++t) bB[t] = loadB(D_ - 4, t);
#pragma unroll
  for (int t = 0; t < 8; ++t) acc[t] = wmma_f32_16x16x4(aA, bA[t], acc[t]);
#pragma unroll
  for (int t = 0; t < 8; ++t) acc[t] = wmma_f32_16x16x4(aB, bB[t], acc[t]);

#pragma unroll
  for (int t = 0; t < 8; ++t)
#pragma unroll
    for (int r = 0; r < 8; ++r) {
      const int m = m0 + r + 8 * (lane >> 4);       // C layout: M = r + 8*(hi half)
      Ch[(size_t)m * LDC + e0 + t * 16 + (lane & 15)] = alpha * acc[t][r];
    }
}

// ---------------------------------------------------------------------------
// Kernel 3: fused scores -> softmax -> attn write -> attn @ vh, head loop,
// per-head partials accumulated in registers (sum over heads, no atomics).
// Block = (b, 32-q-row tile), 256 threads = 8 waves (2 q-halves x 4 e-quarters).
// ---------------------------------------------------------------------------
__global__ void __launch_bounds__(256, 1)
attn_kernel(const float* __restrict__ q0a, const float* __restrict__ q1a,
            const float* __restrict__ k0a, const float* __restrict__ k1a,
            const float* __restrict__ vh,    // [H][B*N][512]
            float* __restrict__ attn_out,    // [B][H][N][N]
            float* __restrict__ outacc) {    // [B*N][512]
  extern __shared__ float smem[];
  float* s_attn = smem;                  // QB * PAD
  float* s_k0 = s_attn + QB * PAD;       // N_
  float* s_k1 = s_k0 + N_;               // N_
  float* s_q0 = s_k1 + N_;               // QB
  float* s_q1 = s_q0 + QB;               // QB
  float* s_rs = s_q1 + QB;               // QB

  const int tid = threadIdx.x;
  const int qb0 = blockIdx.x * QB;
  const int b   = blockIdx.y;
  const int wave = tid >> 5, lane = tid & 31;
  const int qhalf  = wave & 1;           // rows 0-15 / 16-31 of the tile
  const int equart = wave >> 1;          // 128-wide column quarter
  const int arow   = qhalf * 16 + (lane & 15);
  const int khalf  = 2 * (lane >> 4);

  const float* s_arow = s_attn + (size_t)arow * PAD + khalf;

  v8f acc[8] = {};

  for (int h = 0; h < H_; ++h) {
    const size_t bh = ((size_t)b * H_ + h) * N_;
    for (int k = tid; k < N_; k += 256) { s_k0[k] = k0a[bh + k]; s_k1[k] = k1a[bh + k]; }
    if (tid < QB) { s_q0[tid] = q0a[bh + qb0 + tid]; s_q1[tid] = q1a[bh + qb0 + tid]; s_rs[tid] = 0.f; }
    __syncthreads();

    // scores in (0,1] -> exp without a max pass; one LDS atomic per (thread,row)
    for (int q = 0; q < QB; ++q) {
      const float q0 = s_q0[q], q1 = s_q1[q];
      float part = 0.f;
      for (int k = tid; k < N_; k += 256) {
        const float cr = (q0 * s_k1[k]) / (q1 * s_k0[k] + EPSF);
        const float sc = 1.f / (1.f + fabsf(__logf(fabsf(cr) + EPSF)));
        const float ev = __expf(sc);
        s_attn[q * PAD + k] = ev;
        part += ev;
      }
      atomicAdd(&s_rs[q], part);
    }
    __syncthreads();

    // normalize; stream to global attn output, keep tile in LDS for the GEMM
    float* aout = attn_out + (bh + qb0) * (size_t)N_;
    for (int idx = tid; idx < QB * N_; idx += 256) {
      const int q = idx >> 10, k = idx & (N_ - 1);
      const float v = s_attn[q * PAD + k] / s_rs[q];
      s_attn[q * PAD + k] = v;
      aout[(size_t)q * N_ + k] = v;
    }
    __syncthreads();

    // acc += attn_tile[32x1024] @ vh_h[b*N ..][1024x512]  (this wave: 16x128)
    // khalf folded into the base -> all loads are immediate offsets.
    const float* vbl = vh + ((size_t)h * (B_ * N_) + (size_t)b * N_ + khalf) * D_ +
                       equart * 128 + (lane & 15);

    auto loadA = [&](int kb) {
      v2f a; a.x = s_arow[kb]; a.y = s_arow[kb + 1]; return a;
    };
    auto loadB = [&](int kb, int t) {
      const float* bp = vbl + (size_t)kb * D_ + t * 16;
      v2f bv; bv.x = bp[0]; bv.y = bp[D_];   // B[k][e], B[k+1][e]
      return bv;
    };

    v2f aA = loadA(0);
    v2f bA[8];
#pragma unroll
    for (int t = 0; t < 8; ++t) bA[t] = loadB(0, t);
    v2f aB;
    v2f bB[8];

#pragma unroll 1
    for (int kb = 0; kb < N_ - 8; kb += 8) {
      aB = loadA(kb + 4);
#pragma unroll
      for (int t = 0; t < 8; ++t) bB[t] = loadB(kb + 4, t);
      if ((kb & 63) == 0 && kb + 64 < N_)
        __builtin_prefetch(vbl + (size_t)(kb + 64) * D_, 0, 0);  // vh panel ahead
#pragma unroll
      for (int t = 0; t < 8; ++t) acc[t] = wmma_f32_16x16x4(aA, bA[t], acc[t]);
      aA = loadA(kb + 8);
#pragma unroll
      for (int t = 0; t < 8; ++t) bA[t] = loadB(kb + 8, t);
#pragma unroll
      for (int t = 0; t < 8; ++t) acc[t] = wmma_f32_16x16x4(aB, bB[t], acc[t]);
    }
    aB = loadA(N_ - 4);
#pragma unroll
    for (int t = 0; t < 8; ++t) bB[t] = loadB(N_ - 4, t);
#pragma unroll
    for (int t = 0; t < 8; ++t) acc[t] = wmma_f32_16x16x4(aA, bA[t], acc[t]);
#pragma unroll
    for (int t = 0; t < 8; ++t) acc[t] = wmma_f32_16x16x4(aB, bB[t], acc[t]);

    __syncthreads();
  }

  // store sum over heads (block owns these rows exclusively)
#pragma unroll
  for (int t = 0; t < 8; ++t)
#pragma unroll
    for (int r = 0; r < 8; ++r) {
      const int m = b * N_ + qb0 + qhalf * 16 + r + 8 * (lane >> 4);
      outacc[(size_t)m * D_ + equart * 128 + t * 16 + (lane & 15)] = acc[t][r];
    }
}

// ---------------------------------------------------------------------------
extern "C" void kernel_launch(void* const* d_in, const int* in_sizes, int n_in,
                              void* d_out, int out_size, void* d_ws, size_t ws_size,
                              hipStream_t stream) {
  const float* query = (const float*)d_in[0];
  const float* key   = (const float*)d_in[1];
  const float* value = (const float*)d_in[2];
  const float* Wq    = (const float*)d_in[3];
  const float* Wk    = (const float*)d_in[4];
  const float* Wv    = (const float*)d_in[5];
  const float* Wo    = (const float*)d_in[6];

  float* out  = (float*)d_out;                          // [B*N, 512]
  float* attn = out + (size_t)B_ * N_ * D_;             // [B,H,N,N]

  // Workspace layout (~145 MB): vh[H][B*N][512], outacc[B*N][512], qk scalars.
  float* vh     = (float*)d_ws;
  float* outacc = vh + (size_t)H_ * B_ * N_ * D_;
  float* q0a    = outacc + (size_t)B_ * N_ * D_;
  float* q1a    = q0a + (size_t)B_ * H_ * N_;
  float* k0a    = q1a + (size_t)B_ * H_ * N_;
  float* k1a    = k0a + (size_t)B_ * H_ * N_;

  // Constant folds of the projective normalizations (homog coord is exactly 1).
  const float c1    = 1.f / (1.f + EPSF);        // t / (t512+eps), t512 = 1
  const float fqk   = c1 / (c1 + EPSF);          // second normalization in scores
  const float s1    = 1.f / (c1 + EPSF);         // out mean-normalization scale
  const float z512  = c1 * s1;                   // homog coord after Wo
  const float alphaF = (1.f / H_) * s1 * (1.f / (z512 + EPSF));

  qk01_kernel<<<dim3((B_ * H_ * N_) / 8), 256, 0, stream>>>(
      query, key, Wq, Wk, q0a, q1a, k0a, k1a, fqk);

  // vh = c1 * (value @ Wv[h][:512,:512]^T), one z-slice per head
  gemm_xwT_kernel<<<dim3((B_ * N_) / 128, D_ / 128, H_), 256, 0, stream>>>(
      value, 0L, Wv, (long)DP_ * DP_, vh, (long)B_ * N_ * D_, c1);

  const size_t shmem = (QB * PAD + 2 * N_ + 3 * QB) * sizeof(float); // ~140 KB
  attn_kernel<<<dim3(N_ / QB, B_), 256, shmem, stream>>>(
      q0a, q1a, k0a, k1a, vh, attn, outacc);

  // out = alphaF * (sum_h attn@vh) @ Wo[:512,:512]^T  (mean + both norms folded)
  gemm_xwT_kernel<<<dim3((B_ * N_) / 128, D_ / 128, 1), 256, 0, stream>>>(
      outacc, 0L, Wo, 0L, out, 0L, alphaF);
}